// Sheaf_Conv_fixed_66322884984950
// MI455X (gfx1250) — compile-verified
//
#include <hip/hip_runtime.h>

typedef __attribute__((ext_vector_type(2))) float v2f;
typedef __attribute__((ext_vector_type(8))) float v8f;

#define N_NODES   8192
#define E_EDGES   500000
#define ES_USER   495904   /* E - 4096: number of user-path output rows */
#define T_USER    30994    /* ES_USER / 16 */
#define T_TOTAL   31250    /* E / 16 */
#define NB_USER   240
#define NB_ITEM   16
#define NB_TOTAL  (NB_USER + NB_ITEM)

// ---------------------------------------------------------------------------
// Zero the output accumulator (segment_sum baseline).
// ---------------------------------------------------------------------------
__global__ void zero_out_kernel(float* __restrict__ out, int n) {
  int i = blockIdx.x * blockDim.x + threadIdx.x;
  if (i < n) out[i] = 0.0f;
}

// ---------------------------------------------------------------------------
// Precompute fused matrices and biases:
//   M_s[k][n] = sum_y W1[y][k] * W2[y][n]      (s=0: W1=user_w, W2=item_w)
//   cb_s[n]   = sum_y b1[y]   * W2[y][n]       (s=1: W1=item_w, W2=user_w)
// M is stored pre-swizzled into WMMA-B fragment order:
//   fusedM[s*65536 + (kt*16+nt)*64 + lane*2 + p]
//     = M_s[4*kt + 2*(lane>>4) + p][nt*16 + (lane&15)]
// so the main kernel's B load is one contiguous ds_load_b64 per lane.
// ---------------------------------------------------------------------------
__global__ void fuse_kernel(const float* __restrict__ uw, const float* __restrict__ ub,
                            const float* __restrict__ iw, const float* __restrict__ ib,
                            float* __restrict__ fusedM, float* __restrict__ fusedB) {
  int bid = blockIdx.x;
  int n   = threadIdx.x;          // output column 0..255
  if (bid < 512) {
    int s  = bid >> 8;
    int kk = bid & 255;           // input row (k) 0..255
    const float* W1 = s ? iw : uw;
    const float* W2 = s ? uw : iw;
    float acc = 0.0f;
    for (int y = 0; y < 256; ++y)
      acc = fmaf(W1[y * 256 + kk], W2[y * 256 + n], acc);
    int kt = kk >> 2, kr = kk & 3;
    int half = kr >> 1, p = kr & 1;
    int nt = n >> 4, nl = n & 15;
    int lane = half * 16 + nl;
    fusedM[s * 65536 + (kt * 16 + nt) * 64 + lane * 2 + p] = acc;
  } else {
    int s = bid - 512;
    const float* b1 = s ? ib : ub;
    const float* W2 = s ? uw : iw;
    float acc = 0.0f;
    for (int y = 0; y < 256; ++y)
      acc = fmaf(b1[y], W2[y * 256 + n], acc);
    fusedB[s * 256 + n] = acc;
  }
}

// ---------------------------------------------------------------------------
// Main kernel: per 16-edge tile, compute D(16x256) = X(16x256) @ M(256x256)
// with V_WMMA_F32_16X16X4_F32, then out[u[j]] += adj[u[j],v[j]] * (D + cb).
// One 256-thread block = 8 waves; the fused M lives in 256KB LDS, filled via
// GLOBAL_LOAD_ASYNC_TO_LDS_B128 (ASYNCcnt-tracked async copies, no VGPR hop).
// ---------------------------------------------------------------------------
__global__ __launch_bounds__(256, 1) void sheaf_main_kernel(
    const int* __restrict__ ei, const float* __restrict__ emb,
    const float* __restrict__ adj, const float* __restrict__ fusedM,
    const float* __restrict__ fusedB, float* __restrict__ out) {
  extern __shared__ float ldsM[];   // 65536 floats = 256 KB

  int bid  = blockIdx.x;
  int tid  = threadIdx.x;
  int wave = tid >> 5;
  int lane = tid & 31;
  int rowL = lane & 15;             // A-row / N-col this lane owns
  int half = lane >> 4;             // K-half for A/B, M-half for C/D

  int s, tileBeg, tileEnd, gw, nw, srcOff;
  if (bid < NB_USER) {
    s = 0; gw = bid * 8 + wave;            nw = NB_USER * 8;
    tileBeg = 0;       tileEnd = T_USER;   srcOff = 4096;
  } else {
    s = 1; gw = (bid - NB_USER) * 8 + wave; nw = NB_ITEM * 8;
    tileBeg = T_USER;  tileEnd = T_TOTAL;  srcOff = -ES_USER;
  }

  // Async LDS fill: 256 KB, 16B per lane per issue, direct global->LDS DMA.
  // Low 32 bits of the flat shared-pointer address are the LDS byte offset.
  {
    uint32_t ldsBase = (uint32_t)(uintptr_t)ldsM;
    const char* src  = (const char*)(fusedM + s * 65536);
    for (int i = tid; i < 16384; i += 256) {
      uint32_t ldsOff = ldsBase + (uint32_t)i * 16u;
      uint64_t gaddr  = (uint64_t)(uintptr_t)(src + (size_t)i * 16);
      asm volatile("global_load_async_to_lds_b128 %0, %1, off"
                   :: "v"(ldsOff), "v"(gaddr) : "memory");
    }
    asm volatile("s_wait_asynccnt 0x0" ::: "memory");
  }
  __syncthreads();

  // Fused-bias values for each N-tile (column nt*16 + rowL).
  float cb[16];
#pragma unroll
  for (int nt = 0; nt < 16; ++nt) cb[nt] = fusedB[s * 256 + nt * 16 + rowL];

  const float* bbase = ldsM + (size_t)lane * 2;

  for (int tile = tileBeg + gw; tile < tileEnd; tile += nw) {
    int j0 = tile * 16;

    // A-fragment source: embedding row of the *source* edge for this lane's row.
    int jrow    = j0 + rowL;
    int srcEdge = jrow + srcOff;
    int vsrc    = ei[E_EDGES + srcEdge];
    const float* xrow = emb + (size_t)vsrc * 256 + half * 2;

    // Dest-edge metadata for this lane's 8 output rows (m = r + 8*half).
    int   uIdx[8];
    float wv[8];
#pragma unroll
    for (int r = 0; r < 8; ++r) {
      int jm = j0 + r + 8 * half;
      int uu = ei[jm];
      int vv = ei[E_EDGES + jm];
      uIdx[r] = uu;
      wv[r]   = adj[(size_t)uu * 8192 + vv];
    }

    v8f acc[16] = {};

    for (int kt = 0; kt < 64; ++kt) {
      v2f a = *(const v2f*)(xrow + kt * 4);
#pragma unroll
      for (int nt = 0; nt < 16; ++nt) {
        v2f b = *(const v2f*)(bbase + (kt * 16 + nt) * 64);
        acc[nt] = __builtin_amdgcn_wmma_f32_16x16x4_f32(
            /*neg_a=*/false, a, /*neg_b=*/false, b,
            /*c_mod=*/(short)0, acc[nt], /*reuse_a=*/false, /*reuse_b=*/false);
      }
    }

    // Scale by adjacency weight, add fused bias, scatter with f32 atomics.
#pragma unroll
    for (int nt = 0; nt < 16; ++nt) {
      int ncol = nt * 16 + rowL;
#pragma unroll
      for (int r = 0; r < 8; ++r) {
        float val = (acc[nt][r] + cb[nt]) * wv[r];
        unsafeAtomicAdd(out + (size_t)uIdx[r] * 256 + ncol, val);
      }
    }
  }
}

// ---------------------------------------------------------------------------
extern "C" void kernel_launch(void* const* d_in, const int* in_sizes, int n_in,
                              void* d_out, int out_size, void* d_ws, size_t ws_size,
                              hipStream_t stream) {
  const float* adj = (const float*)d_in[0];
  const float* emb = (const float*)d_in[1];
  const int*   ei  = (const int*)d_in[2];
  const float* uw  = (const float*)d_in[3];
  const float* ub  = (const float*)d_in[4];
  const float* iw  = (const float*)d_in[5];
  const float* ib  = (const float*)d_in[6];
  float* out = (float*)d_out;

  float* fusedM = (float*)d_ws;          // 2 * 256*256 floats = 512 KB
  float* fusedB = fusedM + 131072;       // 2 * 256 floats

  int nOut = N_NODES * 256;
  zero_out_kernel<<<(nOut + 255) / 256, 256, 0, stream>>>(out, nOut);
  fuse_kernel<<<514, 256, 0, stream>>>(uw, ub, iw, ib, fusedM, fusedB);
  sheaf_main_kernel<<<NB_TOTAL, 256, 65536 * sizeof(float), stream>>>(
      ei, emb, adj, fusedM, fusedB, out);
}